// LADMM_67302137528440
// MI455X (gfx1250) — compile-verified
//
#include <hip/hip_runtime.h>

#define NSPEC  300
#define NMEAS  9
#define NP     320            // padded n (multiple of 64)
#define BATCH  131072
#define NLAYERS 5
#define AUGW   608            // 304 (left, padded) + 304 (right, padded)

#define BM 128
#define BN 64
#define BK 16
#define LDSP 20               // LDS pitch (BK + 4 floats of padding)

typedef float v2f __attribute__((ext_vector_type(2)));
typedef float v8f __attribute__((ext_vector_type(8)));

__device__ __forceinline__ float softt(float v, float t) {
    return copysignf(fmaxf(fabsf(v) - t, 0.0f), v);
}

// ---------------------------------------------------------------- init state
__global__ void init_state(float* __restrict__ X0, float* __restrict__ ETA,
                           float* __restrict__ TAU, long total) {
    long i = (long)blockIdx.x * blockDim.x + threadIdx.x;
    if (i < total) { X0[i] = 1.0f; ETA[i] = 0.0f; TAU[i] = 0.0f; }
}

// ------------------------------------------- build augmented [M | I] per layer
__global__ void build_aug(const float* __restrict__ A, const float* __restrict__ gamma,
                          const float* __restrict__ alpha, float* __restrict__ AUG) {
    int idx = blockIdx.x * blockDim.x + threadIdx.x;
    const int total = NLAYERS * NSPEC * AUGW;
    if (idx >= total) return;
    int l   = idx / (NSPEC * AUGW);
    int rem = idx % (NSPEC * AUGW);
    int i   = rem / AUGW;
    int j   = rem % AUGW;
    float g = gamma[l], al = alpha[l];
    float val = 0.0f;
    if (j < 304) {                       // left half: M = ATA + al*I + g*DtD
        if (j < NSPEC) {
            float ata = 0.0f;
#pragma unroll
            for (int m = 0; m < NMEAS; ++m) ata += A[m * NSPEC + i] * A[m * NSPEC + j];
            val = ata;
            if (i == j) val += al + 2.0f * g;
            if (j == (i + 1) % NSPEC || i == (j + 1) % NSPEC) val -= g;  // circular neighbors
        }
    } else {                             // right half: identity
        int jr = j - 304;
        val = (jr == i) ? 1.0f : 0.0f;
    }
    AUG[idx] = val;
}

// ------------------------------------- Gauss-Jordan inverse, 1 workgroup/layer
__global__ void gauss_jordan(float* __restrict__ AUG) {
    __shared__ float prow[AUGW];
    __shared__ float pcol[NSPEC];
    float* M = AUG + (long)blockIdx.x * NSPEC * AUGW;
    const int t = threadIdx.x, nt = blockDim.x;
    for (int k = 0; k < NSPEC; ++k) {
        float invp = 1.0f / M[k * AUGW + k];
        for (int j = t; j < AUGW; j += nt)  prow[j] = M[k * AUGW + j] * invp;
        for (int i = t; i < NSPEC; i += nt) pcol[i] = M[i * AUGW + k];
        __syncthreads();
        for (int j = t; j < AUGW; j += nt)  M[k * AUGW + j] = prow[j];
        for (int idx = t; idx < NSPEC * AUGW; idx += nt) {
            int i = idx / AUGW;
            int j = idx % AUGW;
            if (i != k) M[i * AUGW + j] -= pcol[i] * prow[j];
        }
        __syncthreads();
    }
}

// ---------------- extract inverse, fragment-packed for b64 B-loads ----------
// BPK[l][p][col][r] = inv[2p + r][col]  (zero in pad region), p in [0, NP/2)
__global__ void pack_inv(const float* __restrict__ AUG, float* __restrict__ BPK) {
    int idx = blockIdx.x * blockDim.x + threadIdx.x;
    const int total = NLAYERS * NP * NP;
    if (idx >= total) return;
    int l   = idx / (NP * NP);
    int rem = idx % (NP * NP);
    int p   = rem / (2 * NP);
    int q   = rem % (2 * NP);
    int col = q >> 1;
    int r   = q & 1;
    int k   = 2 * p + r;
    float v = 0.0f;
    if (k < NSPEC && col < NSPEC) v = AUG[((long)l * NSPEC + k) * AUGW + 304 + col];
    BPK[idx] = v;
}

// --------------- fused: residual generation (on the fly) + WMMA f32 GEMM ----
// x_new[b,:] = residual[b,:] @ invM ; residual computed from x, eta, tau, b, A
__global__ __launch_bounds__(256) void ladmm_gemm(
    const float* __restrict__ X,   const float* __restrict__ ETA,
    const float* __restrict__ TAU, const float* __restrict__ bvec,
    const float* __restrict__ Amat, const float* __restrict__ BPK,
    const float* __restrict__ gamma, const float* __restrict__ lambda,
    const float* __restrict__ alpha, int layer,
    float* __restrict__ OUT, int ldOut, int ncols)
{
    __shared__ float sres[BM * LDSP];
    const int tid  = threadIdx.x;
    const int wv   = tid >> 5;           // wave id 0..7  -> 16-row slice
    const int lane = tid & 31;
    const int half = lane >> 4;
    const int l16  = lane & 15;
    const int rowBlock = blockIdx.x * BM;
    const int colBlock = blockIdx.y * BN;

    const float g  = gamma[layer], lam = lambda[layer], al = alpha[layer];
    const float ig = 1.0f / g, ial = 1.0f / al, thr = lam / g;

    v8f acc0 = {}, acc1 = {}, acc2 = {}, acc3 = {};

    // staging assignment: thread -> (localRow = tid/2, 8 consecutive k's)
    const int  srow  = tid >> 1;
    const int  skb   = (tid & 1) * 8;
    const long grow  = rowBlock + srow;

    for (int k0 = 0; k0 < NP; k0 += BK) {
#pragma unroll
        for (int kk = 0; kk < 8; ++kk) {
            int k = k0 + skb + kk;
            float res = 0.0f;
            if (k < NSPEC) {
                float xk  = X[grow * NP + k];
                int   km1 = (k == 0) ? (NSPEC - 1) : (k - 1);
                float xm  = X[grow * NP + km1];
                float e   = ETA[grow * NP + k];
                float tv  = TAU[grow * NP + k];
                float ba  = 0.0f;
#pragma unroll
                for (int m = 0; m < NMEAS; ++m)
                    ba += bvec[grow * NMEAS + m] * Amat[m * NSPEC + k];
                float u = softt(xm - xk + e * ig, thr);
                float w = fmaxf(xk + tv * ial, 0.0f);
                res = ba + al * w - tv + g * u - e;
            }
            sres[srow * LDSP + skb + kk] = res;
        }
        __syncthreads();
#pragma unroll
        for (int kk = 0; kk < 4; ++kk) {
            const int kb = kk * 4 + 2 * half;          // K = r + 2*(lane/16)
            v2f a;
            a.x = sres[(wv * 16 + l16) * LDSP + kb + 0];
            a.y = sres[(wv * 16 + l16) * LDSP + kb + 1];
            // packed B: pair index p = kg/2 = k0/2 + 2*kk + half
            const float* Bp = BPK + (long)((k0 >> 1) + 2 * kk + half) * (2 * NP)
                                  + (long)(colBlock + l16) * 2;
            v2f b0 = *(const v2f*)(Bp);
            v2f b1 = *(const v2f*)(Bp + 32);
            v2f b2 = *(const v2f*)(Bp + 64);
            v2f b3 = *(const v2f*)(Bp + 96);
            acc0 = __builtin_amdgcn_wmma_f32_16x16x4_f32(false, a, false, b0, (short)0, acc0, false, false);
            acc1 = __builtin_amdgcn_wmma_f32_16x16x4_f32(false, a, false, b1, (short)0, acc1, false, false);
            acc2 = __builtin_amdgcn_wmma_f32_16x16x4_f32(false, a, false, b2, (short)0, acc2, false, false);
            acc3 = __builtin_amdgcn_wmma_f32_16x16x4_f32(false, a, false, b3, (short)0, acc3, false, false);
        }
        __syncthreads();
    }

    // D layout: VGPR r, lane -> row = r + 8*(lane/16), col = lane%16
    const long rbase = rowBlock + wv * 16 + 8 * half;
    const int  c0    = colBlock + l16;
#pragma unroll
    for (int r = 0; r < 8; ++r) {
        long row = rbase + r;
        if (c0      < ncols) OUT[row * ldOut + c0     ] = acc0[r];
        if (c0 + 16 < ncols) OUT[row * ldOut + c0 + 16] = acc1[r];
        if (c0 + 32 < ncols) OUT[row * ldOut + c0 + 32] = acc2[r];
        if (c0 + 48 < ncols) OUT[row * ldOut + c0 + 48] = acc3[r];
    }
}

// ------------------------------------------------ dual updates (layers 0..3)
__global__ void dual_update(const float* __restrict__ Xold, const float* __restrict__ Xnew,
                            float* __restrict__ ETA, float* __restrict__ TAU,
                            const float* __restrict__ gamma, const float* __restrict__ lambda,
                            const float* __restrict__ alpha, int layer, long total)
{
    long i = (long)blockIdx.x * blockDim.x + threadIdx.x;
    if (i >= total) return;
    long row = i / NSPEC;
    int  j   = (int)(i % NSPEC);
    int  jm1 = (j == 0) ? (NSPEC - 1) : (j - 1);
    float g = gamma[layer], lam = lambda[layer], al = alpha[layer];
    float xo  = Xold[row * NP + j],  xom = Xold[row * NP + jm1];
    float xn  = Xnew[row * NP + j],  xnm = Xnew[row * NP + jm1];
    float e   = ETA[row * NP + j];
    float tv  = TAU[row * NP + j];
    float u = softt(xom - xo + e / g, lam / g);
    float w = fmaxf(xo + tv / al, 0.0f);
    ETA[row * NP + j] = e + g * ((xnm - xn) - u);
    TAU[row * NP + j] = tv + al * (xn - w);
}

// ---------------------------------------------------------------------------
extern "C" void kernel_launch(void* const* d_in, const int* in_sizes, int n_in,
                              void* d_out, int out_size, void* d_ws, size_t ws_size,
                              hipStream_t stream) {
    const float* b      = (const float*)d_in[0];
    const float* A      = (const float*)d_in[2];
    const float* gamma  = (const float*)d_in[3];
    const float* lambda = (const float*)d_in[4];
    const float* alpha  = (const float*)d_in[5];
    float* out = (float*)d_out;
    float* ws  = (float*)d_ws;

    const size_t S = (size_t)BATCH * NP;
    float* X0  = ws;
    float* X1  = ws + S;
    float* ETA = ws + 2 * S;
    float* TAU = ws + 3 * S;
    float* BPK = ws + 4 * S;                                  // 5 * NP * NP (packed inv)
    float* AUG = BPK + (size_t)NLAYERS * NP * NP;             // 5 * 300 * 608

    // state init (every call; no reliance on workspace contents)
    {
        long total = (long)S;
        int blocks = (int)((total + 255) / 256);
        init_state<<<blocks, 256, 0, stream>>>(X0, ETA, TAU, total);
    }
    // build M and invert per layer
    {
        int total = NLAYERS * NSPEC * AUGW;
        build_aug<<<(total + 255) / 256, 256, 0, stream>>>(A, gamma, alpha, AUG);
        gauss_jordan<<<NLAYERS, 512, 0, stream>>>(AUG);
        int t2 = NLAYERS * NP * NP;
        pack_inv<<<(t2 + 255) / 256, 256, 0, stream>>>(AUG, BPK);
    }
    // layers
    const long elems = (long)BATCH * NSPEC;
    for (int l = 0; l < NLAYERS; ++l) {
        const float* Xin = (l % 2 == 0) ? X0 : X1;
        float*       Xout = (l % 2 == 0) ? X1 : X0;
        float* O  = (l == NLAYERS - 1) ? out   : Xout;
        int    ld = (l == NLAYERS - 1) ? NSPEC : NP;
        int    nc = (l == NLAYERS - 1) ? NSPEC : NP;
        dim3 grid(BATCH / BM, NP / BN);
        ladmm_gemm<<<grid, 256, 0, stream>>>(Xin, ETA, TAU, b, A,
                                             BPK + (size_t)l * NP * NP,
                                             gamma, lambda, alpha, l, O, ld, nc);
        if (l < NLAYERS - 1) {
            int blocks = (int)((elems + 255) / 256);
            dual_update<<<blocks, 256, 0, stream>>>(Xin, Xout, ETA, TAU,
                                                    gamma, lambda, alpha, l, elems);
        }
    }
}